// VANILA_GCN_19662360281515
// MI455X (gfx1250) — compile-verified
//
#include <hip/hip_runtime.h>
#include <stdint.h>

#define NNODES 100000
#define NEDGES 3200000
#define IN_DIM 256
#define OUT_DIM 128
#define STATS_BLOCKS 1024

// LDS strides chosen for bank-conflict-free WMMA fragment reads (64 banks):
//  - A tile: 256 rows x 32 k, row stride 36 floats (r*36 mod 64 all-distinct,
//    lane halves offset by +2 dwords -> conflict-free)
//  - W tile: 16 pair-rows x (128 n x 2 k), stride 288 floats (288 mod 64 == 32
//    -> lane halves 0-15 / 16-31 hit disjoint 32-bank groups)
#define LSA_STRIDE 36
#define LSW_STRIDE 288
#define MROWS 256            // rows per block (2 slabs of 16 per wave x 8 waves)

typedef __attribute__((ext_vector_type(2))) float v2f;
typedef __attribute__((ext_vector_type(8))) float v8f;

// ---------------------------------------------------------------------------
// CSR row pointer from sorted COO rows: rowptr[r] = lower_bound(rows, r)
// ---------------------------------------------------------------------------
__global__ void build_rowptr(const int* __restrict__ rows, int* __restrict__ rowptr) {
    int r = blockIdx.x * blockDim.x + threadIdx.x;
    if (r > NNODES) return;
    int lo = 0, hi = NEDGES;
    while (lo < hi) {
        int mid = (lo + hi) >> 1;
        if (rows[mid] < r) lo = mid + 1; else hi = mid;
    }
    rowptr[r] = lo;
}

// ---------------------------------------------------------------------------
// C[M,128] = bn(A)[M,K] @ W[K,128] + bias  using V_WMMA_F32_16X16X4_F32.
// fold==0: A staged via CDNA5 async global->LDS copies (ASYNCcnt path).
// fold==1: A staged manually with per-column (k) BatchNorm scale/shift fused.
// Block: 256 threads (8 waves); each wave owns TWO 16-row M slabs x all 128 N
// so each B fragment (one ds_load_b64) feeds two WMMAs.
// ---------------------------------------------------------------------------
__global__ void __launch_bounds__(256)
gemm_bias_wmma(const float* __restrict__ A, const float* __restrict__ W,
               const float* __restrict__ bias,
               const float* __restrict__ colScale, const float* __restrict__ colShift,
               int fold, float* __restrict__ C, int M, int K) {
    __shared__ float lsA[MROWS * LSA_STRIDE];   // 36,864 B
    __shared__ float lsWp[16 * LSW_STRIDE];     // 18,432 B (k-pair interleaved)

    const int tid   = threadIdx.x;
    const int wave  = tid >> 5;
    const int lane  = tid & 31;
    const int half  = lane >> 4;   // 0: lanes 0-15, 1: lanes 16-31
    const int l16   = lane & 15;
    const int mBase = blockIdx.x * MROWS;
    const uint32_t lsA_base = (uint32_t)(uintptr_t)&lsA[0];

    v8f acc[2][8] = {};

    for (int kc = 0; kc < K; kc += 32) {
        // ---- stage A chunk (256 x 32): rows clamped into valid memory ----
        if (fold) {
            #pragma unroll
            for (int it = 0; it < 8; ++it) {
                int i = tid + it * 256;            // 2048 float4 tiles
                int r = i >> 3;
                int c = (i & 7) << 2;
                int gr = mBase + r; if (gr >= M) gr = M - 1;
                float4 v  = *(const float4*)&A[(size_t)gr * K + kc + c];
                float4 sc = *(const float4*)&colScale[kc + c];
                float4 sh = *(const float4*)&colShift[kc + c];
                v.x = fmaf(v.x, sc.x, sh.x);
                v.y = fmaf(v.y, sc.y, sh.y);
                v.z = fmaf(v.z, sc.z, sh.z);
                v.w = fmaf(v.w, sc.w, sh.w);
                *(float4*)&lsA[r * LSA_STRIDE + c] = v;
            }
        } else {
            #pragma unroll
            for (int it = 0; it < 8; ++it) {
                int i = tid + it * 256;
                int r = i >> 3;
                int c = (i & 7) << 2;
                int gr = mBase + r; if (gr >= M) gr = M - 1;
                const float* gp = &A[(size_t)gr * K + kc + c];
                uint32_t ldst = lsA_base + (uint32_t)((r * LSA_STRIDE + c) * 4);
                asm volatile("global_load_async_to_lds_b128 %0, %1, off"
                             :: "v"(ldst), "v"(gp) : "memory");
            }
        }
        // ---- stage W chunk (32 x 128) into k-pair-interleaved layout ----
        #pragma unroll
        for (int it = 0; it < 2; ++it) {
            int i  = tid + it * 256;               // 512 items
            int p  = i >> 5;                       // pair-row 0..15
            int n0 = (i & 31) << 2;                // n quad
            float4 r0 = *(const float4*)&W[(size_t)(kc + 2 * p    ) * 128 + n0];
            float4 r1 = *(const float4*)&W[(size_t)(kc + 2 * p + 1) * 128 + n0];
            float* dst = &lsWp[p * LSW_STRIDE + n0 * 2];
            *(float4*)(dst + 0) = make_float4(r0.x, r1.x, r0.y, r1.y);
            *(float4*)(dst + 4) = make_float4(r0.z, r1.z, r0.w, r1.w);
        }
        asm volatile("s_wait_asynccnt 0x0" ::: "memory");
        __syncthreads();

        const float* aRow0 = &lsA[(wave * 16 + l16) * LSA_STRIDE];
        const float* aRow1 = aRow0 + 128 * LSA_STRIDE;
        #pragma unroll
        for (int kk = 0; kk < 32; kk += 4) {
            // A frags 16x4 f32: half 0 -> K=kk,kk+1 ; half 1 -> K=kk+2,kk+3
            v2f a0 = *(const v2f*)&aRow0[kk + half * 2];
            v2f a1 = *(const v2f*)&aRow1[kk + half * 2];
            const int p = (kk >> 1) + half;        // interleaved pair-row
            #pragma unroll
            for (int nt = 0; nt < 8; ++nt) {
                int n = nt * 16 + l16;
                v2f b = *(const v2f*)&lsWp[p * LSW_STRIDE + n * 2];  // one ds_load_b64
                acc[0][nt] = __builtin_amdgcn_wmma_f32_16x16x4_f32(
                    false, a0, false, b, (short)0, acc[0][nt], false, false);
                acc[1][nt] = __builtin_amdgcn_wmma_f32_16x16x4_f32(
                    false, a1, false, b, (short)0, acc[1][nt], false, false);
            }
        }
        __syncthreads();
    }

    // Epilogue: C/D layout -> VGPR i: lanes0-15 row i, lanes16-31 row i+8
    #pragma unroll
    for (int s = 0; s < 2; ++s) {
        #pragma unroll
        for (int nt = 0; nt < 8; ++nt) {
            int n = nt * 16 + l16;
            float bv = bias[n];
            #pragma unroll
            for (int i = 0; i < 8; ++i) {
                int row = mBase + s * 128 + wave * 16 + half * 8 + i;
                if (row < M) C[(size_t)row * 128 + n] = acc[s][nt][i] + bv;
            }
        }
    }
}

// ---------------------------------------------------------------------------
// out[r,:] = relu( sum_e vals[e] * h[cols[e],:] )  one 128-thread block / row.
// Edge scalars are wave-uniform -> scalar loads; gathers are coalesced 512B
// lines resident in the 192MB L2 (h is 51.2MB). 4-way unrolled accumulators.
// ---------------------------------------------------------------------------
__global__ void __launch_bounds__(128)
spmm_relu(const float* __restrict__ h, const int* __restrict__ rowptr,
          const int* __restrict__ cols, const float* __restrict__ vals,
          float* __restrict__ out) {
    int r = blockIdx.x;
    int d = threadIdx.x;
    int e0 = rowptr[r];
    int e1 = rowptr[r + 1];
    float a0 = 0.f, a1 = 0.f, a2 = 0.f, a3 = 0.f;
    int e = e0;
    for (; e + 4 <= e1; e += 4) {
        int   c0 = cols[e],   c1 = cols[e + 1], c2 = cols[e + 2], c3 = cols[e + 3];
        float v0 = vals[e],   v1 = vals[e + 1], v2 = vals[e + 2], v3 = vals[e + 3];
        a0 = fmaf(v0, h[(size_t)c0 * 128 + d], a0);
        a1 = fmaf(v1, h[(size_t)c1 * 128 + d], a1);
        a2 = fmaf(v2, h[(size_t)c2 * 128 + d], a2);
        a3 = fmaf(v3, h[(size_t)c3 * 128 + d], a3);
    }
    for (; e < e1; ++e)
        a0 = fmaf(vals[e], h[(size_t)cols[e] * 128 + d], a0);
    out[(size_t)r * 128 + d] = fmaxf((a0 + a1) + (a2 + a3), 0.f);
}

// ---------------------------------------------------------------------------
// Deterministic two-pass column stats -> fused scale/shift for BN
// ---------------------------------------------------------------------------
__global__ void __launch_bounds__(128)
colstats_partial(const float* __restrict__ x, float* __restrict__ part) {
    int d = threadIdx.x;
    float s = 0.f, s2 = 0.f;
    for (int r = blockIdx.x; r < NNODES; r += gridDim.x) {
        float v = x[(size_t)r * 128 + d];
        s  += v;
        s2 += v * v;
    }
    part[blockIdx.x * 256 + d]       = s;
    part[blockIdx.x * 256 + 128 + d] = s2;
}

__global__ void __launch_bounds__(128)
colstats_final(const float* __restrict__ part,
               const float* __restrict__ gamma, const float* __restrict__ beta,
               float* __restrict__ scale, float* __restrict__ shift) {
    int d = threadIdx.x;
    float s = 0.f, s2 = 0.f;
    for (int b = 0; b < STATS_BLOCKS; ++b) {
        s  += part[b * 256 + d];
        s2 += part[b * 256 + 128 + d];
    }
    float mean = s / (float)NNODES;
    float var  = s2 / (float)NNODES - mean * mean;
    float inv  = 1.0f / sqrtf(var + 1e-9f);
    float sc   = gamma[d] * inv;
    scale[d] = sc;
    shift[d] = beta[d] - mean * sc;
}

// out = x * scale[d] + shift[d]   (final BN apply, float4)
__global__ void __launch_bounds__(256)
bn_apply(const float4* __restrict__ x, const float* __restrict__ scale,
         const float* __restrict__ shift, float4* __restrict__ out) {
    int i = blockIdx.x * blockDim.x + threadIdx.x;   // over NNODES*32 float4
    if (i >= NNODES * 32) return;
    int d4 = (i & 31) << 2;
    float4 v  = x[i];
    float4 sc = *(const float4*)&scale[d4];
    float4 sh = *(const float4*)&shift[d4];
    v.x = fmaf(v.x, sc.x, sh.x);
    v.y = fmaf(v.y, sc.y, sh.y);
    v.z = fmaf(v.z, sc.z, sh.z);
    v.w = fmaf(v.w, sc.w, sh.w);
    out[i] = v;
}

// ---------------------------------------------------------------------------
extern "C" void kernel_launch(void* const* d_in, const int* in_sizes, int n_in,
                              void* d_out, int out_size, void* d_ws, size_t ws_size,
                              hipStream_t stream) {
    const float* f_in = (const float*)d_in[0];
    const int*   rows = (const int*)d_in[1];
    const int*   cols = (const int*)d_in[2];
    const float* vals = (const float*)d_in[3];
    const float* W0   = (const float*)d_in[4];
    const float* b0   = (const float*)d_in[5];
    const float* W1   = (const float*)d_in[6];
    const float* b1   = (const float*)d_in[7];
    const float* g0   = (const float*)d_in[8];
    const float* be0  = (const float*)d_in[9];
    const float* g1   = (const float*)d_in[10];
    const float* be1  = (const float*)d_in[11];
    float* out = (float*)d_out;

    // workspace layout (offsets 128B aligned)
    char* ws = (char*)d_ws;
    const size_t HBYTES = (size_t)NNODES * 128 * sizeof(float);     // 51,200,000
    float* bufA   = (float*)(ws);
    float* bufB   = (float*)(ws + HBYTES);
    int*   rowptr = (int*)  (ws + 2 * HBYTES);                      // N+1 ints
    float* part   = (float*)(ws + 2 * HBYTES + 400128);             // 1024*256 f
    float* scalev = (float*)(ws + 2 * HBYTES + 400128 + 1048576);
    float* shiftv = (float*)(ws + 2 * HBYTES + 400128 + 1048576 + 512);

    const int gemmBlocks = (NNODES + MROWS - 1) / MROWS;   // 391
    const int bnBlocks   = (NNODES * 32 + 255) / 256;

    build_rowptr<<<(NNODES + 256) / 256, 256, 0, stream>>>(rows, rowptr);

    // ---- layer 0 : GEMM (async A staging) -> SpMM+ReLU -> stats ----
    gemm_bias_wmma<<<gemmBlocks, 256, 0, stream>>>(f_in, W0, b0,
                                                   (const float*)nullptr,
                                                   (const float*)nullptr,
                                                   0, bufA, NNODES, IN_DIM);
    spmm_relu<<<NNODES, 128, 0, stream>>>(bufA, rowptr, cols, vals, bufB);
    colstats_partial<<<STATS_BLOCKS, 128, 0, stream>>>(bufB, part);
    colstats_final<<<1, 128, 0, stream>>>(part, g0, be0, scalev, shiftv);

    // ---- layer 1 : BN0 folded into GEMM A staging -> SpMM+ReLU -> BN1 ----
    gemm_bias_wmma<<<gemmBlocks, 256, 0, stream>>>(bufB, W1, b1,
                                                   scalev, shiftv,
                                                   1, bufA, NNODES, OUT_DIM);
    spmm_relu<<<NNODES, 128, 0, stream>>>(bufA, rowptr, cols, vals, bufB);
    colstats_partial<<<STATS_BLOCKS, 128, 0, stream>>>(bufB, part);
    colstats_final<<<1, 128, 0, stream>>>(part, g1, be1, scalev, shiftv);
    bn_apply<<<bnBlocks, 256, 0, stream>>>((const float4*)bufB, scalev, shiftv, (float4*)out);
}